// Rotation_14199161880927
// MI455X (gfx1250) — compile-verified
//
#include <hip/hip_runtime.h>
#include <math.h>

// Problem constants (match the reference).
#define IMG_H 3072
#define IMG_W 3072
#define IMG_C 3

// ---------------------------------------------------------------------------
// Single fused kernel: one thread per output pixel.
//   block = 256 threads  -> 8 wave32 waves, 256 consecutive columns of a row
//   grid  = (W/256, H)
//
// theta = deg2rad(frame_value[0] * 20) is in [0, 0.3491] rad, so sin/cos are
// evaluated with short Taylor polynomials (abs err < 3e-10, far below fp32
// resolution) instead of libm's Payne-Hanek path (~400 SALU ops). This also
// lets us drop the separate prologue kernel: one launch, no serialized
// dependency on a 1-thread kernel in a ~10us memory-bound job.
//
// Stores: 3 contiguous floats per lane -> lanes tile a contiguous 3 KB run,
// emitted non-temporal (output is streamed once; keep the reused input image
// resident in the 192 MB L2, which holds the whole 113 MB input).
// Loads: 4 taps x 3 contiguous floats; consecutive lanes step ~cos(theta)*12B
// through the source row -> coalesced; taps shared between neighbors hit L0/L2.
// ---------------------------------------------------------------------------
__global__ __launch_bounds__(256) void rot_bilinear_kernel(
    const float* __restrict__ img,          // (H, W, 3)
    const float* __restrict__ frame_value,  // 1 element
    float* __restrict__ out)                // (H, W, 3)
{
    const int c = blockIdx.x * 256 + threadIdx.x;   // output column
    const int r = blockIdx.y;                       // output row
    if (c >= IMG_W) return;

    // ---- angle & rotation matrix (uniform; stays in scalar/cheap VALU) ----
    // theta = frame_value[0] * 20 deg = frame_value[0] * 0.34906585 rad
    const float t  = frame_value[0] * 0.34906585039886593f;
    const float t2 = t * t;
    // sin(t) = t*(1 - t^2/6 + t^4/120 - t^6/5040)         |err| < 2.2e-10
    const float st = t * fmaf(t2, fmaf(t2, fmaf(t2, -1.9841270e-4f,
                                                     8.3333333e-3f),
                                            -1.6666667e-1f), 1.0f);
    // cos(t) = 1 - t^2/2 + t^4/24 - t^6/720 + t^8/40320   |err| < 1e-11
    const float ct = fmaf(t2, fmaf(t2, fmaf(t2, fmaf(t2, 2.4801587e-5f,
                                                         -1.3888889e-3f),
                                                 4.1666667e-2f),
                                        -0.5f), 1.0f);

    // keras transform_matrix_offset_center: rotate about (dim/2 + 0.5)
    const float o_r = (float)IMG_H * 0.5f + 0.5f;
    const float o_c = (float)IMG_W * 0.5f + 0.5f;

    const float dr = (float)r - o_r;
    const float dc = (float)c - o_c;

    // src = R @ (out - o) + o, R = [[cos,-sin],[sin,cos]]
    float src_r = fmaf(ct, dr, fmaf(-st, dc, o_r));
    float src_c = fmaf(st, dr, fmaf(ct, dc, o_c));

    // fill_mode='nearest' -> clamp to valid range (edge replicate)
    src_r = fminf(fmaxf(src_r, 0.0f), (float)(IMG_H - 1));
    src_c = fminf(fmaxf(src_c, 0.0f), (float)(IMG_W - 1));

    const float r0f = floorf(src_r);
    const float c0f = floorf(src_c);
    const float wr = src_r - r0f;
    const float wc = src_c - c0f;

    const int r0 = (int)r0f;
    const int c0 = (int)c0f;
    const int r1 = min(r0 + 1, IMG_H - 1);
    const int c1 = min(c0 + 1, IMG_W - 1);

    const float w00 = (1.0f - wr) * (1.0f - wc);
    const float w01 = (1.0f - wr) * wc;
    const float w10 = wr * (1.0f - wc);
    const float w11 = wr * wc;

    // 4 taps, 3 contiguous floats each (backend merges to b96-class loads).
    const float* __restrict__ p00 = img + ((size_t)r0 * IMG_W + c0) * IMG_C;
    const float* __restrict__ p01 = img + ((size_t)r0 * IMG_W + c1) * IMG_C;
    const float* __restrict__ p10 = img + ((size_t)r1 * IMG_W + c0) * IMG_C;
    const float* __restrict__ p11 = img + ((size_t)r1 * IMG_W + c1) * IMG_C;

    const float a0 = p00[0], a1 = p00[1], a2 = p00[2];
    const float b0 = p01[0], b1 = p01[1], b2 = p01[2];
    const float g0 = p10[0], g1 = p10[1], g2 = p10[2];
    const float h0 = p11[0], h1 = p11[1], h2 = p11[2];

    const float acc0 = fmaf(w00, a0, fmaf(w01, b0, fmaf(w10, g0, w11 * h0)));
    const float acc1 = fmaf(w00, a1, fmaf(w01, b1, fmaf(w10, g1, w11 * h1)));
    const float acc2 = fmaf(w00, a2, fmaf(w01, b2, fmaf(w10, g2, w11 * h2)));

    // Streamed-once output: non-temporal stores (TH=NT) so the output stream
    // does not evict the (reused, L2-resident) input image.
    float* __restrict__ po = out + ((size_t)r * IMG_W + c) * IMG_C;
    __builtin_nontemporal_store(acc0, po + 0);
    __builtin_nontemporal_store(acc1, po + 1);
    __builtin_nontemporal_store(acc2, po + 2);
}

// ---------------------------------------------------------------------------
// Harness entry point.
//   d_in[0] = x           (1*3072*3072*3 float32)
//   d_in[1] = frame_value (1 float32)
//   d_out   = (1,3072,3072,3) float32
// ---------------------------------------------------------------------------
extern "C" void kernel_launch(void* const* d_in, const int* in_sizes, int n_in,
                              void* d_out, int out_size, void* d_ws, size_t ws_size,
                              hipStream_t stream) {
    (void)in_sizes; (void)n_in; (void)out_size; (void)d_ws; (void)ws_size;

    const float* img = (const float*)d_in[0];
    const float* fv  = (const float*)d_in[1];
    float* out       = (float*)d_out;

    dim3 block(256);
    dim3 grid(IMG_W / 256, IMG_H);
    rot_bilinear_kernel<<<grid, block, 0, stream>>>(img, fv, out);
}